// SAGENet_58454504898645
// MI455X (gfx1250) — compile-verified
//
#include <hip/hip_runtime.h>
#include <math.h>

typedef float v2f __attribute__((ext_vector_type(2)));
typedef float v8f __attribute__((ext_vector_type(8)));

#define N_NODES 100000
#define N_EDGES 3200000
#define F_IN    128
#define HIDDEN  64
#define NCLS    7
#define ROW_TILES (N_NODES / 16)   // 6250, exact

// ---------------- utility ----------------
__global__ void fill_f32(float* __restrict__ p, int n, float v) {
    int i = blockIdx.x * blockDim.x + threadIdx.x;
    if (i < n) p[i] = v;
}

// Wcat[128,128]: rows 0..63 = W1_l (64x128), rows 64..127 = W1_r
__global__ void build_wcat(const float* __restrict__ Wl,
                           const float* __restrict__ Wr,
                           float* __restrict__ Wcat) {
    int i = blockIdx.x * blockDim.x + threadIdx.x;      // 0..16383
    if (i >= 128 * 128) return;
    int r = i >> 7, c = i & 127;
    Wcat[i] = (r < 64) ? Wl[r * 128 + c] : Wr[(r - 64) * 128 + c];
}

// Wp[16,64]: rows 0..6 = W2_l, rows 8..14 = W2_r, rows 7,15 = 0
__global__ void build_wp(const float* __restrict__ W2l,
                         const float* __restrict__ W2r,
                         float* __restrict__ Wp) {
    int i = blockIdx.x * blockDim.x + threadIdx.x;      // 0..1023
    if (i >= 16 * 64) return;
    int r = i >> 6, c = i & 63;
    float v = 0.0f;
    if (r < 7)                 v = W2l[r * 64 + c];
    else if (r >= 8 && r < 15) v = W2r[(r - 8) * 64 + c];
    Wp[i] = v;
}

// ---------------- fp32 WMMA GEMM:  Y[N, n_jt*16] = X[N,K] @ W^T ----------------
// W is [n_jt*16, K]. One wave computes one 16x16 output tile using
// V_WMMA_F32_16X16X4_F32, looping K in steps of 4.
// A 16x4 layout: lanes 0-15 -> (M=lane, K=0/1 in v0/v1); lanes 16-31 -> K=2/3.
// B 4x16 layout: mirrored (N=lane&15, K per half).  C/D: v[i] = D[M = i + 8*half][N = lane&15].
__global__ void wmma_gemm_xt(const float* __restrict__ X,
                             const float* __restrict__ W,
                             float* __restrict__ Y,
                             int K, int ldy, int n_jt) {
    int wave = threadIdx.x >> 5;
    int lane = threadIdx.x & 31;
    int gw   = blockIdx.x * (blockDim.x >> 5) + wave;   // global wave id (uniform per wave)
    int tile = gw / n_jt;
    int jt   = gw % n_jt;
    if (tile >= ROW_TILES) return;                      // wave-uniform guard: EXEC stays all-1s

    int half = lane >> 4;
    int l15  = lane & 15;
    const float* xrow = X + (size_t)(tile * 16 + l15) * K;
    const float* wrow = W + (size_t)(jt   * 16 + l15) * K;

    v8f c = {};
    for (int k0 = 0; k0 < K; k0 += 4) {
        v2f a = *(const v2f*)(xrow + k0 + half * 2);
        v2f b = *(const v2f*)(wrow + k0 + half * 2);
        // (neg_a, A, neg_b, B, c_mod, C, reuse_a, reuse_b)
        c = __builtin_amdgcn_wmma_f32_16x16x4_f32(false, a, false, b,
                                                  (short)0, c, false, false);
    }

    float* yout = Y + (size_t)(tile * 16) * ldy + jt * 16 + l15;
#pragma unroll
    for (int i = 0; i < 8; ++i)
        yout[(size_t)(i + half * 8) * ldy] = c[i];
}

// ---------------- degree: cnt[dst] += 1 (one thread per edge) ----------------
__global__ void degree_k(const int* __restrict__ ei, float* __restrict__ cnt) {
    int e = blockIdx.x * blockDim.x + threadIdx.x;
    if (e >= N_EDGES) return;
    atomicAdd(&cnt[ei[N_EDGES + e]], 1.0f);
}

// ---------------- layer-1 edge scatter: agg1[dst] += xl[src] ----------------
// xl = columns 0..63 of Y1 (row stride 128). One thread per (edge, 4-float chunk):
// float4 gather (global_load_b128) + 4 f32 atomics.
__global__ void scatter1(const int* __restrict__ ei,
                         const float* __restrict__ Y1,
                         float* __restrict__ agg1) {
    int t = blockIdx.x * blockDim.x + threadIdx.x;      // E*16 = 51,200,000 threads
    int e = t >> 4;
    int q = t & 15;
    if (e >= N_EDGES) return;
    int s = ei[e];
    int d = ei[N_EDGES + e];
    const float4 v = *(const float4*)(Y1 + (size_t)s * 128 + q * 4);
    float* dst = agg1 + (size_t)d * 64 + q * 4;
    atomicAdd(dst + 0, v.x);
    atomicAdd(dst + 1, v.y);
    atomicAdd(dst + 2, v.z);
    atomicAdd(dst + 3, v.w);
}

// ---------------- h = relu(agg1/max(cnt,1) + xr + b1), in place over agg1 ----------------
__global__ void relu_combine(float* __restrict__ agg1,
                             const float* __restrict__ Y1,
                             const float* __restrict__ cnt,
                             const float* __restrict__ b1) {
    int t = blockIdx.x * blockDim.x + threadIdx.x;      // N*16 threads (float4 each)
    if (t >= N_NODES * 16) return;
    int n = t >> 4, q = t & 15;
    float cdiv = fmaxf(cnt[n], 1.0f);
    float rinv = 1.0f / cdiv;
    float4 a  = *(float4*)(agg1 + (size_t)n * 64 + q * 4);
    float4 xr = *(const float4*)(Y1 + (size_t)n * 128 + 64 + q * 4);
    float4 bb = *(const float4*)(b1 + q * 4);
    float4 r;
    r.x = fmaxf(a.x * rinv + xr.x + bb.x, 0.0f);
    r.y = fmaxf(a.y * rinv + xr.y + bb.y, 0.0f);
    r.z = fmaxf(a.z * rinv + xr.z + bb.z, 0.0f);
    r.w = fmaxf(a.w * rinv + xr.w + bb.w, 0.0f);
    *(float4*)(agg1 + (size_t)n * 64 + q * 4) = r;
}

// ---------------- layer-2 edge scatter: agg2[dst][0..6] += P[src][0..6] ----------------
// P row stride 16; cols 0..6 = hl, col 7 = 0 (padded), cols 8..14 = hr.
// One thread per (edge, half): chunk 0 -> f 0..3, chunk 1 -> f 4..6 (f=7 is always 0).
__global__ void scatter2(const int* __restrict__ ei,
                         const float* __restrict__ P,
                         float* __restrict__ agg2) {
    int t = blockIdx.x * blockDim.x + threadIdx.x;      // E*2 = 6,400,000 threads
    int e = t >> 1;
    int q = t & 1;
    if (e >= N_EDGES) return;
    int s = ei[e];
    int d = ei[N_EDGES + e];
    const float4 v = *(const float4*)(P + (size_t)s * 16 + q * 4);
    float* dst = agg2 + (size_t)d * 8 + q * 4;
    atomicAdd(dst + 0, v.x);
    atomicAdd(dst + 1, v.y);
    atomicAdd(dst + 2, v.z);
    if (q == 0) atomicAdd(dst + 3, v.w);                // skip f=7 (always 0)
}

// ---------------- out = log_softmax(agg2/max(cnt,1) + hr + b2) ----------------
__global__ void final_k(const float* __restrict__ agg2,
                        const float* __restrict__ P,
                        const float* __restrict__ cnt,
                        const float* __restrict__ b2,
                        float* __restrict__ out) {
    int n = blockIdx.x * blockDim.x + threadIdx.x;
    if (n >= N_NODES) return;
    float cdiv = fmaxf(cnt[n], 1.0f);
    float rinv = 1.0f / cdiv;
    float o[NCLS];
    float m = -3.0e38f;
#pragma unroll
    for (int f = 0; f < NCLS; ++f) {
        float v = agg2[(size_t)n * 8 + f] * rinv + P[(size_t)n * 16 + 8 + f] + b2[f];
        o[f] = v;
        m = fmaxf(m, v);
    }
    float s = 0.0f;
#pragma unroll
    for (int f = 0; f < NCLS; ++f) s += expf(o[f] - m);
    float l = logf(s);
#pragma unroll
    for (int f = 0; f < NCLS; ++f) out[(size_t)n * NCLS + f] = o[f] - m - l;
}

// ---------------- launch ----------------
extern "C" void kernel_launch(void* const* d_in, const int* in_sizes, int n_in,
                              void* d_out, int out_size, void* d_ws, size_t ws_size,
                              hipStream_t stream) {
    const float* x    = (const float*)d_in[0];
    const int*   ei   = (const int*)  d_in[1];
    const float* W1l  = (const float*)d_in[2];
    const float* W1r  = (const float*)d_in[3];
    const float* b1   = (const float*)d_in[4];
    const float* W2l  = (const float*)d_in[5];
    const float* W2r  = (const float*)d_in[6];
    const float* b2   = (const float*)d_in[7];
    float* out = (float*)d_out;

    // workspace layout (floats)
    float* ws   = (float*)d_ws;
    float* Wcat = ws;                                    //   128*128      = 16,384
    float* Y1   = Wcat + 128 * 128;                      //   N*128        = 12,800,000
    float* agg1 = Y1   + (size_t)N_NODES * 128;          //   N*64         =  6,400,000
    float* cnt  = agg1 + (size_t)N_NODES * 64;           //   N            =    100,000
    float* Wp   = cnt  + N_NODES;                        //   16*64        =      1,024
    float* P    = Wp   + 16 * 64;                        //   N*16         =  1,600,000
    float* agg2 = P    + (size_t)N_NODES * 16;           //   N*8          =    800,000
    // total ~87 MB

    // 1) Wcat = [W1_l ; W1_r]
    build_wcat<<<64, 256, 0, stream>>>(W1l, W1r, Wcat);

    // 2) Y1 = x @ Wcat^T   (N x 128; cols 0..63 = xl, 64..127 = xr)
    wmma_gemm_xt<<<ROW_TILES, 256, 0, stream>>>(x, Wcat, Y1, 128, 128, 8);

    // 3) zero agg1 + cnt (contiguous)
    {
        int n = N_NODES * 64 + N_NODES;                  // 6,500,000
        fill_f32<<<(n + 255) / 256, 256, 0, stream>>>(agg1, n, 0.0f);
    }

    // 4) degree counts + scatter xl over edges (float4 gathers)
    degree_k<<<N_EDGES / 256, 256, 0, stream>>>(ei, cnt);
    scatter1<<<(N_EDGES * 16) / 256, 256, 0, stream>>>(ei, Y1, agg1);

    // 5) h = relu(agg1/cnt + xr + b1), in place in agg1
    relu_combine<<<(N_NODES * 16 + 255) / 256, 256, 0, stream>>>(agg1, Y1, cnt, b1);

    // 6) Wp = [W2_l ; 0 ; W2_r ; 0] padded to 16 rows
    build_wp<<<4, 256, 0, stream>>>(W2l, W2r, Wp);

    // 7) P = h @ Wp^T   (N x 16)
    wmma_gemm_xt<<<(ROW_TILES + 7) / 8, 256, 0, stream>>>(agg1, Wp, P, 64, 16, 1);

    // 8) zero agg2
    {
        int n = N_NODES * 8;
        fill_f32<<<(n + 255) / 256, 256, 0, stream>>>(agg2, n, 0.0f);
    }

    // 9) scatter hl over edges (7 atomics/edge, float4 gathers)
    scatter2<<<(N_EDGES * 2) / 256, 256, 0, stream>>>(ei, P, agg2);

    // 10) log_softmax(agg2/cnt + hr + b2)
    final_k<<<(N_NODES + 255) / 256, 256, 0, stream>>>(agg2, P, cnt, b2, out);
}